// LocalSelfAttentionWithSEAndCCA_61263413510146
// MI455X (gfx1250) — compile-verified
//
#include <hip/hip_runtime.h>
#include <hip/hip_bf16.h>

typedef __attribute__((ext_vector_type(2))) float v2f;
typedef __attribute__((ext_vector_type(8))) float v8f;

namespace {
constexpr int BATCH = 4;
constexpr int CCH   = 256;   // Cout / Cin
constexpr int LEN   = 2048;
constexpr int HEADS = 8;
constexpr int HDIM  = 32;
constexpr int KW    = 9;
constexpr int PADW  = 4;
constexpr int CH2   = 128;   // Cout/2 for CCA hidden
constexpr int RSE   = 16;    // SE bottleneck
constexpr int NTOT  = BATCH * LEN;  // 8192 columns
constexpr float EPS = 1e-5f;
}

// ---------------------------------------------------------------------------
// fp32 WMMA GEMM:  D[M x N] = W[M x K] * X[K x N]   (N = BATCH*LEN)
// Column n = (b, l); activation source position is l+shift (zero if OOB),
// implementing 1x1 convs (shift=0) and conv3 taps (shift=-1,0,+1).
// xLayout: 0 -> X element (c, b, pos) at X[(b*K + c)*LEN + pos]  (original x)
//          1 -> at X[(b*LEN + pos)*K + c]                        (our (b,l,c))
// D stored (b,l,o): D[n*M + o]. accumulate!=0 -> D +=.
//
// Per-wave 32x32 output (2x2 tiles of 16x16, 4 x v8f accumulators) so each
// A/B fragment pair is reused by two WMMAs: 4 WMMAs per 4 LDS fragment loads.
// Block tile 64(M) x 128(N), 8 waves, K staged in chunks of 32.
// ---------------------------------------------------------------------------
__global__ __launch_bounds__(256) void gemm_wmma(
    const float* __restrict__ W, int wRowStride, int wColStride,
    const float* __restrict__ X, int xLayout,
    float* __restrict__ D,
    int M, int Kdim, int shift, int accumulate)
{
  __shared__ float Ws[32][65];    // Ws[k][m], block M-tile = 64
  __shared__ float Xs[32][129];   // Xs[k][n], block N-tile = 128
  const int tid  = threadIdx.x;
  const int lane = tid & 31;
  const int wave = tid >> 5;
  const int n0 = blockIdx.x * 128;
  const int m0 = blockIdx.y * 64;
  const int mW = (wave & 1) * 32;   // wave M offset within block tile
  const int nW = (wave >> 1) * 32;  // wave N offset within block tile

  v8f acc00 = {0.f,0.f,0.f,0.f,0.f,0.f,0.f,0.f};
  v8f acc01 = acc00, acc10 = acc00, acc11 = acc00;

  const int half = lane >> 4;       // 0: lanes 0-15, 1: lanes 16-31
  const int mr   = lane & 15;

  for (int k0 = 0; k0 < Kdim; k0 += 32) {
    // ---- stage W chunk (64 m x 32 k), 8 elems/thread, k contiguous ----
    #pragma unroll
    for (int i = 0; i < 8; ++i) {
      int idx = i * 256 + tid;
      int kj = idx & 31;
      int mi = idx >> 5;
      Ws[kj][mi] = W[(size_t)(m0 + mi) * wRowStride + (size_t)(k0 + kj) * wColStride];
    }
    // ---- stage X chunk (32 k x 128 n), 16 elems/thread ----
    if (xLayout == 1) {
      #pragma unroll
      for (int i = 0; i < 16; ++i) {
        int idx = i * 256 + tid;
        int kj = idx & 31, nj = idx >> 5;
        int n = n0 + nj;
        int b = n / LEN, l = n % LEN;
        int pos = l + shift;
        float v = 0.f;
        if ((unsigned)pos < (unsigned)LEN)
          v = X[((size_t)b * LEN + pos) * Kdim + (k0 + kj)];
        Xs[kj][nj] = v;
      }
    } else {
      #pragma unroll
      for (int i = 0; i < 16; ++i) {
        int idx = i * 256 + tid;
        int nj = idx & 127, kj = idx >> 7;
        int n = n0 + nj;
        int b = n / LEN, l = n % LEN;
        int pos = l + shift;
        float v = 0.f;
        if ((unsigned)pos < (unsigned)LEN)
          v = X[((size_t)b * Kdim + (k0 + kj)) * LEN + pos];
        Xs[kj][nj] = v;
      }
    }
    __syncthreads();

    // ---- 8 k-steps x 4 WMMAs (V_WMMA_F32_16X16X4_F32) ----
    #pragma unroll
    for (int ks = 0; ks < 32; ks += 4) {
      int kb = ks + half * 2;
      v2f a0, a1, b0, b1;
      // A 16x4: VGPR0 = K{0,2}, VGPR1 = K{1,3}  (lanes 0-15 / 16-31)
      a0[0] = Ws[kb][mW + mr];        a0[1] = Ws[kb + 1][mW + mr];
      a1[0] = Ws[kb][mW + 16 + mr];   a1[1] = Ws[kb + 1][mW + 16 + mr];
      // B 4x16: same striping, row across lanes
      b0[0] = Xs[kb][nW + mr];        b0[1] = Xs[kb + 1][nW + mr];
      b1[0] = Xs[kb][nW + 16 + mr];   b1[1] = Xs[kb + 1][nW + 16 + mr];
      acc00 = __builtin_amdgcn_wmma_f32_16x16x4_f32(false, a0, false, b0, (short)0, acc00, false, false);
      acc01 = __builtin_amdgcn_wmma_f32_16x16x4_f32(false, a0, false, b1, (short)0, acc01, false, false);
      acc10 = __builtin_amdgcn_wmma_f32_16x16x4_f32(false, a1, false, b0, (short)0, acc10, false, false);
      acc11 = __builtin_amdgcn_wmma_f32_16x16x4_f32(false, a1, false, b1, (short)0, acc11, false, false);
    }
    __syncthreads();
  }

  // ---- store D (C/D layout: VGPR v -> row v (lanes 0-15) / v+8 (16-31)) ----
  const int colBase = n0 + nW + (lane & 15);
  const int rowBase = m0 + mW + (lane >> 4) * 8;
  #pragma unroll
  for (int im = 0; im < 2; ++im) {
    #pragma unroll
    for (int in = 0; in < 2; ++in) {
      const v8f& a = im == 0 ? (in == 0 ? acc00 : acc01) : (in == 0 ? acc10 : acc11);
      size_t col = colBase + in * 16;
      int row = rowBase + im * 16;
      #pragma unroll
      for (int v = 0; v < 8; ++v) {
        size_t di = col * M + (row + v);
        if (accumulate) D[di] += a[v]; else D[di] = a[v];
      }
    }
  }
}

// ---------------------------------------------------------------------------
// Local attention: one wave per (b,h,l); lane = channel-in-head.
// OOB window slot: k = rel only (padded kf == wk@0 == 0), v = 0.
// ---------------------------------------------------------------------------
__global__ __launch_bounds__(256) void attn_kernel(
    const float* __restrict__ Q, const float* __restrict__ Kf,
    const float* __restrict__ Vf, const float* __restrict__ rel,
    float* __restrict__ Out)
{
  int wid  = blockIdx.x * 8 + (threadIdx.x >> 5);
  int lane = threadIdx.x & 31;
  int b = wid / (HEADS * LEN);
  int rem = wid % (HEADS * LEN);
  int h = rem / LEN;
  int l = rem % LEN;
  int c = h * HDIM + lane;

  float q = Q[((size_t)b * LEN + l) * CCH + c];

  float scores[KW];
  #pragma unroll
  for (int k = 0; k < KW; ++k) {
    int pos = l - PADW + k;
    float kv = rel[lane * KW + k];
    if ((unsigned)pos < (unsigned)LEN)
      kv += Kf[((size_t)b * LEN + pos) * CCH + c];
    float p = q * kv;
    #pragma unroll
    for (int m = 16; m >= 1; m >>= 1) p += __shfl_xor(p, m, 32);
    scores[k] = p * 0.17677669529663687f;  // 1/sqrt(32)
  }
  float mx = scores[0];
  #pragma unroll
  for (int k = 1; k < KW; ++k) mx = fmaxf(mx, scores[k]);
  float w[KW];
  float den = 0.f;
  #pragma unroll
  for (int k = 0; k < KW; ++k) { w[k] = __expf(scores[k] - mx); den += w[k]; }
  float inv = 1.f / den;
  float accv = 0.f;
  #pragma unroll
  for (int k = 0; k < KW; ++k) {
    int pos = l - PADW + k;
    if ((unsigned)pos < (unsigned)LEN)
      accv += w[k] * Vf[((size_t)b * LEN + pos) * CCH + c];
  }
  Out[((size_t)b * LEN + l) * CCH + c] = accv * inv;
}

// Per-channel sum / sumsq over all (b,l): one block per channel.
__global__ __launch_bounds__(256) void stats_kernel(
    const float* __restrict__ X, int Cdim,
    float* __restrict__ sums, float* __restrict__ sqs)
{
  int c = blockIdx.x;
  float s = 0.f, s2 = 0.f;
  for (int i = threadIdx.x; i < NTOT; i += 256) {
    float v = X[(size_t)i * Cdim + c];
    s += v; s2 += v * v;
  }
  __shared__ float sh[256], sh2[256];
  sh[threadIdx.x] = s; sh2[threadIdx.x] = s2;
  __syncthreads();
  for (int st = 128; st >= 1; st >>= 1) {
    if (threadIdx.x < st) {
      sh[threadIdx.x]  += sh[threadIdx.x + st];
      sh2[threadIdx.x] += sh2[threadIdx.x + st];
    }
    __syncthreads();
  }
  if (threadIdx.x == 0) { sums[c] = sh[0]; sqs[c] = sh2[0]; }
}

// y = relu(bn(x))  (training-mode stats from sums/sqs)
__global__ __launch_bounds__(256) void bnrelu_kernel(
    const float* __restrict__ X, int Cdim,
    const float* __restrict__ sums, const float* __restrict__ sqs,
    const float* __restrict__ g, const float* __restrict__ bb,
    float* __restrict__ Y)
{
  size_t idx = (size_t)blockIdx.x * 256 + threadIdx.x;
  int c = (int)(idx % Cdim);
  float m   = sums[c] * (1.f / NTOT);
  float var = sqs[c] * (1.f / NTOT) - m * m;
  float sc  = g[c] * rsqrtf(var + EPS);
  float sh  = bb[c] - m * sc;
  Y[idx] = fmaxf(X[idx] * sc + sh, 0.f);
}

// Per-(b,c) sums over L (for SE pooling): one block per (b,c).
__global__ __launch_bounds__(256) void bcsum_kernel(
    const float* __restrict__ Z, float* __restrict__ sbc)
{
  int bc = blockIdx.x;
  int b = bc / CCH, c = bc % CCH;
  float s = 0.f;
  for (int l = threadIdx.x; l < LEN; l += 256)
    s += Z[((size_t)b * LEN + l) * CCH + c];
  __shared__ float sh[256];
  sh[threadIdx.x] = s;
  __syncthreads();
  for (int st = 128; st >= 1; st >>= 1) {
    if (threadIdx.x < st) sh[threadIdx.x] += sh[threadIdx.x + st];
    __syncthreads();
  }
  if (threadIdx.x == 0) sbc[bc] = sh[0];
}

// SE MLP (tiny): gates[b,c] = sigmoid(relu(bn2_pool @ w1^T + b1) @ w2^T + b2)
__global__ __launch_bounds__(256) void se_kernel(
    const float* __restrict__ zsum, const float* __restrict__ zsq,
    const float* __restrict__ sbc,
    const float* __restrict__ g2, const float* __restrict__ b2,
    const float* __restrict__ w1, const float* __restrict__ b1,
    const float* __restrict__ w2, const float* __restrict__ b2se,
    float* __restrict__ gates)
{
  __shared__ float s_sh[BATCH][CCH];
  __shared__ float hid[BATCH][RSE];
  int t = threadIdx.x;  // == channel
  float m   = zsum[t] * (1.f / NTOT);
  float var = zsq[t] * (1.f / NTOT) - m * m;
  float sc  = g2[t] * rsqrtf(var + EPS);
  float shf = b2[t] - m * sc;
  for (int b = 0; b < BATCH; ++b)
    s_sh[b][t] = (sbc[b * CCH + t] * (1.f / LEN)) * sc + shf;
  __syncthreads();
  if (t < BATCH * RSE) {
    int b = t / RSE, r = t % RSE;
    float a = b1[r];
    for (int c = 0; c < CCH; ++c) a += s_sh[b][c] * w1[r * CCH + c];
    hid[b][r] = fmaxf(a, 0.f);
  }
  __syncthreads();
  for (int b = 0; b < BATCH; ++b) {
    float a = b2se[t];
    #pragma unroll
    for (int r = 0; r < RSE; ++r) a += hid[b][r] * w2[t * RSE + r];
    gates[b * CCH + t] = 1.f / (1.f + __expf(-a));
  }
}

// OUT = bn2(Z) * se_gate   (this "out" feeds CCA and final product)
__global__ __launch_bounds__(256) void outgate_kernel(
    const float* __restrict__ Z,
    const float* __restrict__ zsum, const float* __restrict__ zsq,
    const float* __restrict__ g2, const float* __restrict__ b2,
    const float* __restrict__ gates, float* __restrict__ Out)
{
  size_t idx = (size_t)blockIdx.x * 256 + threadIdx.x;
  int c = (int)(idx % CCH);
  int b = (int)(idx / ((size_t)LEN * CCH));
  float m   = zsum[c] * (1.f / NTOT);
  float var = zsq[c] * (1.f / NTOT) - m * m;
  float sc  = g2[c] * rsqrtf(var + EPS);
  float shf = b2[c] - m * sc;
  Out[idx] = (Z[idx] * sc + shf) * gates[b * CCH + c];
}

// d_out[b,c,l] = OUT * sigmoid(bn(A2))   (transpose (b,l,c) -> (b,c,l))
__global__ __launch_bounds__(256) void final_kernel(
    const float* __restrict__ OutT, const float* __restrict__ A2,
    const float* __restrict__ a2sum, const float* __restrict__ a2sq,
    const float* __restrict__ g, const float* __restrict__ bb,
    float* __restrict__ dout)
{
  size_t idx = (size_t)blockIdx.x * 256 + threadIdx.x;  // (b,c,l), l fastest
  int l = (int)(idx % LEN);
  int c = (int)((idx / LEN) % CCH);
  int b = (int)(idx / ((size_t)CCH * LEN));
  size_t src = ((size_t)b * LEN + l) * CCH + c;
  float m   = a2sum[c] * (1.f / NTOT);
  float var = a2sq[c] * (1.f / NTOT) - m * m;
  float sc  = g[c] * rsqrtf(var + EPS);
  float shf = bb[c] - m * sc;
  float gate = 1.f / (1.f + __expf(-(A2[src] * sc + shf)));
  dout[idx] = OutT[src] * gate;
}

// ---------------------------------------------------------------------------
extern "C" void kernel_launch(void* const* d_in, const int* in_sizes, int n_in,
                              void* d_out, int out_size, void* d_ws, size_t ws_size,
                              hipStream_t stream) {
  const float* x    = (const float*)d_in[0];
  const float* rel  = (const float*)d_in[1];
  const float* wq   = (const float*)d_in[2];
  const float* wk   = (const float*)d_in[3];
  const float* wv   = (const float*)d_in[4];
  const float* aggw = (const float*)d_in[5];
  const float* ag1  = (const float*)d_in[6];
  const float* ab1  = (const float*)d_in[7];
  const float* ag2  = (const float*)d_in[8];
  const float* ab2  = (const float*)d_in[9];
  const float* sew1 = (const float*)d_in[10];
  const float* seb1 = (const float*)d_in[11];
  const float* sew2 = (const float*)d_in[12];
  const float* seb2 = (const float*)d_in[13];
  const float* cw1  = (const float*)d_in[14];
  const float* cg1  = (const float*)d_in[15];
  const float* cb1  = (const float*)d_in[16];
  const float* cw2  = (const float*)d_in[17];
  const float* cg2  = (const float*)d_in[18];
  const float* cb2  = (const float*)d_in[19];

  // Workspace layout (floats). Activations are 2,097,152 floats (8 MB) each;
  // total ws used ~34 MB. Regions are reused once dead.
  float* ws  = (float*)d_ws;
  float* Q   = ws;
  float* Kf  = ws + 2097152;
  float* Vf  = ws + 4194304;
  float* ATT = ws + 6291456;
  float* T   = Q;    // BN1+ReLU output (Q dead after attention)
  float* Z   = Kf;   // agg GEMM output (Kf dead after attention)
  float* OUT = Vf;   // gated "out"     (Vf dead after attention)
  float* A1  = ATT;  // CCA hidden      (ATT dead after bnrelu)
  float* A1N = Q;    // CCA hidden post-BN (T dead after agg GEMM)
  float* A2  = Kf;   // CCA out         (Z dead after outgate)
  float* S   = ws + 8388608;
  float* attSum = S + 0,    *attSq = S + 256;
  float* zSum   = S + 512,  *zSq   = S + 768;
  float* sbc    = S + 1024;
  float* gates  = S + 2048;
  float* a1Sum  = S + 3072, *a1Sq  = S + 3200;
  float* a2Sum  = S + 3328, *a2Sq  = S + 3584;

  dim3 blk(256);
  dim3 g256(NTOT / 128, CCH / 64);   // M=256 GEMMs: 64 x 4 blocks
  dim3 g128(NTOT / 128, CH2 / 64);   // M=128 GEMM (conv1): 64 x 2 blocks

  // 1) q/k/v projections (fp32 WMMA GEMMs), outputs in (b,l,c)
  gemm_wmma<<<g256, blk, 0, stream>>>(wq, CCH, 1, x, 0, Q,  CCH, CCH, 0, 0);
  gemm_wmma<<<g256, blk, 0, stream>>>(wk, CCH, 1, x, 0, Kf, CCH, CCH, 0, 0);
  gemm_wmma<<<g256, blk, 0, stream>>>(wv, CCH, 1, x, 0, Vf, CCH, CCH, 0, 0);

  // 2) local windowed attention (VALU; ~75 MFLOP)
  attn_kernel<<<BATCH * HEADS * LEN / 8, blk, 0, stream>>>(Q, Kf, Vf, rel, ATT);

  // 3) agg: BN1 -> ReLU -> 1x1 GEMM
  stats_kernel<<<CCH, blk, 0, stream>>>(ATT, CCH, attSum, attSq);
  bnrelu_kernel<<<(NTOT * CCH) / 256, blk, 0, stream>>>(ATT, CCH, attSum, attSq, ag1, ab1, T);
  gemm_wmma<<<g256, blk, 0, stream>>>(aggw, CCH, 1, T, 1, Z, CCH, CCH, 0, 0);

  // 4) BN2 stats + SE pooling + SE MLP + gate
  stats_kernel<<<CCH, blk, 0, stream>>>(Z, CCH, zSum, zSq);
  bcsum_kernel<<<BATCH * CCH, blk, 0, stream>>>(Z, sbc);
  se_kernel<<<1, blk, 0, stream>>>(zSum, zSq, sbc, ag2, ab2, sew1, seb1, sew2, seb2, gates);
  outgate_kernel<<<(NTOT * CCH) / 256, blk, 0, stream>>>(Z, zSum, zSq, ag2, ab2, gates, OUT);

  // 5) CCA conv1 (k=3, 256->128) as 3 shifted accumulating WMMA GEMMs
  for (int t = 0; t < 3; ++t)
    gemm_wmma<<<g128, blk, 0, stream>>>(cw1 + t, CCH * 3, 3, OUT, 1, A1, CH2, CCH, t - 1, t > 0);
  stats_kernel<<<CH2, blk, 0, stream>>>(A1, CH2, a1Sum, a1Sq);
  bnrelu_kernel<<<(NTOT * CH2) / 256, blk, 0, stream>>>(A1, CH2, a1Sum, a1Sq, cg1, cb1, A1N);

  // 6) CCA conv2 (k=3, 128->256) as 3 shifted accumulating WMMA GEMMs
  for (int t = 0; t < 3; ++t)
    gemm_wmma<<<g256, blk, 0, stream>>>(cw2 + t, CH2 * 3, 3, A1N, 1, A2, CCH, CH2, t - 1, t > 0);
  stats_kernel<<<CCH, blk, 0, stream>>>(A2, CCH, a2Sum, a2Sq);

  // 7) final: d_out = OUT * sigmoid(bn(A2)), transposed back to (B,C,L)
  final_kernel<<<(size_t)(BATCH * CCH * LEN) / 256, blk, 0, stream>>>(
      OUT, A2, a2Sum, a2Sq, cg2, cb2, (float*)d_out);
}